// SAGE_lr_84954453114989
// MI455X (gfx1250) — compile-verified
//
#include <hip/hip_runtime.h>
#include <hip/hip_bf16.h>
#include <stdint.h>

#define N_NODES 100000
#define N_EDGES 1600000
#define DIN     128

typedef float v2f __attribute__((ext_vector_type(2)));
typedef float v8f __attribute__((ext_vector_type(8)));

// ---------------------------------------------------------------- utilities

__global__ void zero_kernel(float4* __restrict__ p, int n4) {
  int i = blockIdx.x * blockDim.x + threadIdx.x;
  if (i < n4) p[i] = float4{0.f, 0.f, 0.f, 0.f};
}

__global__ void degree_kernel(const int* __restrict__ dst,
                              float* __restrict__ deg, int E) {
  int e = blockIdx.x * blockDim.x + threadIdx.x;
  if (e < E) {
    __hip_atomic_fetch_add(&deg[dst[e]], 1.0f,
                           __ATOMIC_RELAXED, __HIP_MEMORY_SCOPE_AGENT);
  }
}

__global__ void invdeg_kernel(const float* __restrict__ deg,
                              float* __restrict__ inv, int n) {
  int i = blockIdx.x * blockDim.x + threadIdx.x;
  if (i < n) {
    float d = deg[i];
    inv[i] = 1.0f / (d > 1.0f ? d : 1.0f);
  }
}

// WT[k][n] = W[n][k]  (W is [dout][din] row-major) -> coalesced B-fragment loads
__global__ void transpose_kernel(const float* __restrict__ W,
                                 float* __restrict__ WT, int dout, int din) {
  int idx = blockIdx.x * blockDim.x + threadIdx.x;
  if (idx < dout * din) {
    int n = idx / din;
    int k = idx - n * din;
    WT[k * dout + n] = W[idx];
  }
}

// ------------------------------------------------- edge scatter (mean-sum)
// One wave32 per edge; lane l moves channels [4l, 4l+4). 32*4 = 128 channels.
__global__ __launch_bounds__(256) void scatter_kernel(
    const float* __restrict__ feat, const int* __restrict__ src,
    const int* __restrict__ dst, float* __restrict__ agg, int E) {
  int gid  = blockIdx.x * blockDim.x + threadIdx.x;
  int e    = gid >> 5;
  int lane = gid & 31;
  if (e >= E) return;
  size_t s = (size_t)src[e] * DIN;
  size_t d = (size_t)dst[e] * DIN;
  float4 v = *(const float4*)(feat + s + lane * 4);
  float* base = agg + d + lane * 4;
  __hip_atomic_fetch_add(base + 0, v.x, __ATOMIC_RELAXED, __HIP_MEMORY_SCOPE_AGENT);
  __hip_atomic_fetch_add(base + 1, v.y, __ATOMIC_RELAXED, __HIP_MEMORY_SCOPE_AGENT);
  __hip_atomic_fetch_add(base + 2, v.z, __ATOMIC_RELAXED, __HIP_MEMORY_SCOPE_AGENT);
  __hip_atomic_fetch_add(base + 3, v.w, __ATOMIC_RELAXED, __HIP_MEMORY_SCOPE_AGENT);
}

// ---------------------------------------------- fused SAGE GEMM (+ReLU+LN)
// Block: 16 nodes, DOUT/16 waves; wave w computes the 16x16 tile for output
// columns [16w, 16w+16).  D = (agg*invdeg) @ Wl^T + X @ Wr^T + b.
// FP32 WMMA 16x16x4, K-loop over DIN=128.
template <int DOUT, bool LN>
__global__ __launch_bounds__(DOUT * 2) void sage_gemm_kernel(
    const float* __restrict__ X, const float* __restrict__ AGG,
    const float* __restrict__ INV,
    const float* __restrict__ WlT,   // [DIN][DOUT]
    const float* __restrict__ WrT,   // [DIN][DOUT]
    const float* __restrict__ bias,
    const float* __restrict__ gamma, const float* __restrict__ beta,
    float* __restrict__ OUT, int nN) {
  __shared__ float sX[16][DIN + 4];
  __shared__ float sA[16][DIN + 4];
  __shared__ float sO[16][DOUT + 4];
  __shared__ float sMu[16];
  __shared__ float sRs[16];

  const int base = blockIdx.x * 16;
  const int tid  = threadIdx.x;

  // Cooperative tile stage: x rows + invdeg-scaled aggregate rows.
  for (int idx = tid; idx < 16 * DIN; idx += DOUT * 2) {
    int r = idx >> 7;            // DIN == 128
    int c = idx & (DIN - 1);
    int node = base + r;
    if (node >= nN) node = nN - 1;     // clamp (N divisible by 16 anyway)
    size_t off = (size_t)node * DIN + c;
    sX[r][c] = X[off];
    sA[r][c] = AGG[off] * INV[node];
  }
  __syncthreads();

  const int wave = tid >> 5;
  const int lane = tid & 31;
  const int hi   = lane >> 4;    // which half-wave
  const int lm   = lane & 15;
  const int n    = wave * 16 + lm;   // output column of this lane's tile

  __builtin_prefetch(&WlT[n], 0, 1);
  __builtin_prefetch(&WrT[n], 0, 1);

  v8f acc = {0.f, 0.f, 0.f, 0.f, 0.f, 0.f, 0.f, 0.f};

#if __has_builtin(__builtin_amdgcn_wmma_f32_16x16x4_f32)
  // A fragment (16x4 f32): lane -> M = lane&15; VGPR j -> K = kb + 2*hi + j
  // B fragment (4x16 f32): lane -> N = lane&15; VGPR j -> K = kb + 2*hi + j
#pragma unroll 4
  for (int kb = 0; kb < DIN; kb += 4) {
    const int k0 = kb + 2 * hi;
    v2f aA, aX, bL, bR;
    aA[0] = sA[lm][k0];  aA[1] = sA[lm][k0 + 1];
    aX[0] = sX[lm][k0];  aX[1] = sX[lm][k0 + 1];
    bL[0] = WlT[(size_t)k0 * DOUT + n];
    bL[1] = WlT[(size_t)(k0 + 1) * DOUT + n];
    bR[0] = WrT[(size_t)k0 * DOUT + n];
    bR[1] = WrT[(size_t)(k0 + 1) * DOUT + n];
    acc = __builtin_amdgcn_wmma_f32_16x16x4_f32(
        false, aA, false, bL, (short)0, acc, false, false);
    acc = __builtin_amdgcn_wmma_f32_16x16x4_f32(
        false, aX, false, bR, (short)0, acc, false, false);
  }
#else
  // Scalar fallback with identical D-fragment semantics.
  for (int v = 0; v < 8; ++v) {
    int m = v + 8 * hi;
    float s = 0.f;
    for (int k = 0; k < DIN; ++k)
      s += sA[m][k] * WlT[(size_t)k * DOUT + n] +
           sX[m][k] * WrT[(size_t)k * DOUT + n];
    acc[v] = s;
  }
#endif

  // D layout: VGPR v -> row M = v + 8*hi, col N = wave*16 + (lane&15)
  const float bv = bias[n];

  if (LN) {
#pragma unroll
    for (int v = 0; v < 8; ++v) {
      float val = acc[v] + bv;
      sO[v + 8 * hi][n] = val > 0.f ? val : 0.f;   // ReLU
    }
    __syncthreads();
    if (tid < 16) {
      float s = 0.f, s2 = 0.f;
      for (int c = 0; c < DOUT; ++c) {
        float v = sO[tid][c];
        s += v; s2 += v * v;
      }
      float mu  = s * (1.0f / DOUT);
      float var = s2 * (1.0f / DOUT) - mu * mu;
      sMu[tid] = mu;
      sRs[tid] = rsqrtf(var + 1e-5f);
    }
    __syncthreads();
    for (int idx = tid; idx < 16 * DOUT; idx += DOUT * 2) {
      int r = idx / DOUT;
      int c = idx - r * DOUT;
      int node = base + r;
      if (node < nN) {
        float v = (sO[r][c] - sMu[r]) * sRs[r] * gamma[c] + beta[c];
        OUT[(size_t)node * DOUT + c] = v;
      }
    }
  } else {
#pragma unroll
    for (int v = 0; v < 8; ++v) {
      int node = base + v + 8 * hi;
      if (node < nN) OUT[(size_t)node * DOUT + n] = acc[v] + bv;
    }
  }
}

// ------------------------------------------------------------------ driver

extern "C" void kernel_launch(void* const* d_in, const int* in_sizes, int n_in,
                              void* d_out, int out_size, void* d_ws, size_t ws_size,
                              hipStream_t stream) {
  const float* x   = (const float*)d_in[0];
  const int*   ei  = (const int*)d_in[1];   // [2, E] (JAX default x64-off -> int32)
  const int*   src = ei;
  const int*   dst = ei + N_EDGES;
  const float* Wl0 = (const float*)d_in[2];
  const float* Wr0 = (const float*)d_in[3];
  const float* b0  = (const float*)d_in[4];
  const float* Wl1 = (const float*)d_in[5];
  const float* Wr1 = (const float*)d_in[6];
  const float* b1  = (const float*)d_in[7];
  const float* Wl2 = (const float*)d_in[8];
  const float* Wr2 = (const float*)d_in[9];
  const float* b2  = (const float*)d_in[10];
  const float* g0  = (const float*)d_in[11];
  const float* be0 = (const float*)d_in[12];
  const float* g1  = (const float*)d_in[13];
  const float* be1 = (const float*)d_in[14];
  float* out = (float*)d_out;

  // Workspace carve-up (floats).
  float* w      = (float*)d_ws;
  float* invdeg = w;  w += ((N_NODES + 63) & ~63);
  float* WlT0   = w;  w += 128 * 128;
  float* WrT0   = w;  w += 128 * 128;
  float* WlT1   = w;  w += 128 * 128;
  float* WrT1   = w;  w += 128 * 128;
  float* WlT2   = w;  w += 128 * 64;
  float* WrT2   = w;  w += 128 * 64;
  float* agg    = w;  w += (size_t)N_NODES * DIN;
  float* h0     = w;  w += (size_t)N_NODES * DIN;
  float* h1     = w;

  const int AGG4 = N_NODES * DIN / 4;   // float4 count of agg buffer

  // Degree / inverse degree (deg temporarily lives in agg buffer).
  zero_kernel<<<(N_NODES / 4 + 255) / 256, 256, 0, stream>>>((float4*)agg, N_NODES / 4);
  degree_kernel<<<(N_EDGES + 255) / 256, 256, 0, stream>>>(dst, agg, N_EDGES);
  invdeg_kernel<<<(N_NODES + 255) / 256, 256, 0, stream>>>(agg, invdeg, N_NODES);

  // Weight transposes.
  transpose_kernel<<<(128 * 128 + 255) / 256, 256, 0, stream>>>(Wl0, WlT0, 128, 128);
  transpose_kernel<<<(128 * 128 + 255) / 256, 256, 0, stream>>>(Wr0, WrT0, 128, 128);
  transpose_kernel<<<(128 * 128 + 255) / 256, 256, 0, stream>>>(Wl1, WlT1, 128, 128);
  transpose_kernel<<<(128 * 128 + 255) / 256, 256, 0, stream>>>(Wr1, WrT1, 128, 128);
  transpose_kernel<<<(64 * 128 + 255) / 256, 256, 0, stream>>>(Wl2, WlT2, 64, 128);
  transpose_kernel<<<(64 * 128 + 255) / 256, 256, 0, stream>>>(Wr2, WrT2, 64, 128);

  const int scatterBlocks = (N_EDGES * 32 + 255) / 256;
  const int gemmBlocks    = (N_NODES + 15) / 16;

  // Layer 0: x -> h0 (ReLU + LN)
  zero_kernel<<<(AGG4 + 255) / 256, 256, 0, stream>>>((float4*)agg, AGG4);
  scatter_kernel<<<scatterBlocks, 256, 0, stream>>>(x, src, dst, agg, N_EDGES);
  sage_gemm_kernel<128, true><<<gemmBlocks, 256, 0, stream>>>(
      x, agg, invdeg, WlT0, WrT0, b0, g0, be0, h0, N_NODES);

  // Layer 1: h0 -> h1 (ReLU + LN)
  zero_kernel<<<(AGG4 + 255) / 256, 256, 0, stream>>>((float4*)agg, AGG4);
  scatter_kernel<<<scatterBlocks, 256, 0, stream>>>(h0, src, dst, agg, N_EDGES);
  sage_gemm_kernel<128, true><<<gemmBlocks, 256, 0, stream>>>(
      h0, agg, invdeg, WlT1, WrT1, b1, g1, be1, h1, N_NODES);

  // Layer 2: h1 -> out (no activation)
  zero_kernel<<<(AGG4 + 255) / 256, 256, 0, stream>>>((float4*)agg, AGG4);
  scatter_kernel<<<scatterBlocks, 256, 0, stream>>>(h1, src, dst, agg, N_EDGES);
  sage_gemm_kernel<64, false><<<gemmBlocks, 128, 0, stream>>>(
      h1, agg, invdeg, WlT2, WrT2, b2, nullptr, nullptr, out, N_NODES);
}